// GNNEncoder_5153960755630
// MI455X (gfx1250) — compile-verified
//
#include <hip/hip_runtime.h>
#include <hip/hip_bf16.h>

typedef __attribute__((ext_vector_type(2))) float v2f;
typedef __attribute__((ext_vector_type(8))) float v8f;

#define NNODES 50000
#define DIN    64
#define HID    128
#define DOUT   64

// ---------------------------------------------------------------------------
// CDNA5 async global->LDS copy (ASYNCcnt-tracked) + wait helpers.
// The dynamic __shared__ array is the only LDS object in the kernel, so its
// base offset is 0 and raw byte offsets address it directly.
// ---------------------------------------------------------------------------
__device__ __forceinline__ void async_copy16(unsigned ldsByteOff, const float* g) {
    asm volatile("global_load_async_to_lds_b128 %0, %1, off"
                 :: "v"(ldsByteOff), "v"((unsigned long long)(uintptr_t)g)
                 : "memory");
}
template <int N>
__device__ __forceinline__ void wait_asynccnt() {
    asm volatile("s_wait_asynccnt %0" :: "i"(N) : "memory");
}

// ---------------------------------------------------------------------------
// Edge-degree count: cnt[dst[e]] += 1.0f   (computed once, reused both layers)
// ---------------------------------------------------------------------------
__global__ void count_edges_kernel(const long long* __restrict__ dst,
                                   float* __restrict__ cnt, int E) {
    int e = blockIdx.x * blockDim.x + threadIdx.x;
    if (e < E) atomicAdd(&cnt[(int)dst[e]], 1.0f);
}

// ---------------------------------------------------------------------------
// Scatter-add of gathered features: agg[dst[e], :] += x[src[e], :]
// One thread per (edge, float4 chunk); 16/32 consecutive lanes share an edge
// -> coalesced gather + coalesced atomics, all L2-resident (<=25.6MB tables).
// ---------------------------------------------------------------------------
__global__ void scatter_add4_kernel(const float* __restrict__ x,
                                    const long long* __restrict__ src,
                                    const long long* __restrict__ dst,
                                    float* __restrict__ agg, int E, int d) {
    const int q = d >> 2;
    long long gid = (long long)blockIdx.x * blockDim.x + threadIdx.x;
    long long total = (long long)E * q;
    if (gid >= total) return;
    int e = (int)(gid / q);
    int f = ((int)(gid % q)) << 2;
    int s  = (int)src[e];
    int dd = (int)dst[e];
    const float4 v = *(const float4*)(x + (size_t)s * d + f);
    float* o = agg + (size_t)dd * d + f;
    atomicAdd(o + 0, v.x);
    atomicAdd(o + 1, v.y);
    atomicAdd(o + 2, v.z);
    atomicAdd(o + 3, v.w);
}

// ---------------------------------------------------------------------------
// Fused SAGE GEMM, fp32 WMMA (V_WMMA_F32_16X16X4_F32):
//   HAS_AGG=1: out = [relu]( (agg/max(cnt,1)) @ Wl + X @ Wr + bias )
//   HAS_AGG=0: out = X @ Wr + bias
//
// Weights staged into LDS in double-buffered 16-row K-chunks via
// global_load_async_to_lds_b128 / s_wait_asynccnt, shared by all 8 waves of
// the block. Staged rows are padded by 8 floats (LDW = NOUT+8) so the two
// K-row groups of a wave (rows k and k+2, bank offset 2*LDW mod 64 = 16)
// hit disjoint LDS bank sets -> conflict-free ds reads.
//
// Wave tile: 16(M) x 64(N). All waves stage + barrier; only valid waves
// (m0 < N, wave-uniform) compute/store, so EXEC is all-1s for every WMMA
// and barriers never deadlock.
//
// ISA VGPR layouts (CDNA5 ISA):
//   A 16x4 f32 : M = lane&15 ; VGPR0/1 = K {0,1} (lanes 0-15) / {2,3} (16-31)
//   B  4x16 f32: N = lane&15 ; VGPR0/1 = K {0,1} (lanes 0-15) / {2,3} (16-31)
//   C 16x16 f32: VGPR i -> M = i + (lane>=16 ? 8:0), N = lane&15
// ---------------------------------------------------------------------------
template <int NOUT, int K, bool HAS_AGG>
__global__ void sage_gemm_wmma_kernel(const float* __restrict__ agg,
                                      const float* __restrict__ cnt,
                                      const float* __restrict__ X,
                                      const float* __restrict__ Wl,
                                      const float* __restrict__ Wr,
                                      const float* __restrict__ bias,
                                      float* __restrict__ out,
                                      int N, int doRelu) {
    constexpr int CK         = 16;                    // K rows per staged chunk
    constexpr int LDW        = NOUT + 8;              // padded LDS row stride
    constexpr int NMAT       = HAS_AGG ? 2 : 1;       // matrices staged
    constexpr int COLT       = NOUT / 64;             // col tiles per row tile
    constexpr int BROWS      = (8 / COLT) * 16;       // rows per block
    constexpr int MAT_FLOATS = CK * NOUT;             // unpadded chunk of one W
    constexpr int BUF_LDS    = NMAT * CK * LDW;       // padded LDS floats / buf
    constexpr int NCHUNK     = K / CK;
    constexpr int PER_THREAD = (NMAT * MAT_FLOATS / 4) / 256; // async ops/thread

    extern __shared__ float smem[];                   // [2][BUF_LDS]

    const int tid  = threadIdx.x;
    const int lane = tid & 31;
    const int w    = tid >> 5;

    const int m0 = blockIdx.x * BROWS + (w / COLT) * 16;
    const int j0 = (w % COLT) * 64;
    const bool valid = (m0 < N);                      // wave-uniform

    const int r   = lane & 15;
    const int hiK = (lane >> 4) << 1;                 // 0 or 2

    float scale = 1.0f;
    if (HAS_AGG && valid) scale = 1.0f / fmaxf(cnt[m0 + r], 1.0f);

    v8f c0 = {}, c1 = {}, c2 = {}, c3 = {};

    auto issue_chunk = [&](int c, int buf) {
        const unsigned ldsBase = (unsigned)(buf * BUF_LDS * 4);
#pragma unroll
        for (int i = 0; i < PER_THREAD; ++i) {
            const int fl  = (i * 256 + tid) * 4;      // float idx, NMAT chunks
            const int mat = (HAS_AGG && fl >= MAT_FLOATS) ? 1 : 0;
            const int flm = fl - mat * MAT_FLOATS;
            const int row = flm / NOUT;
            const int col = flm % NOUT;
            const float* W = (HAS_AGG && mat == 0) ? Wl : Wr;
            const float* g = W + (size_t)c * MAT_FLOATS + flm;
            const unsigned off = ldsBase + (unsigned)((mat * CK + row) * LDW + col) * 4u;
            async_copy16(off, g);
        }
    };

    issue_chunk(0, 0);
#pragma unroll
    for (int c = 0; c < NCHUNK; ++c) {
        if (c + 1 < NCHUNK) {
            issue_chunk(c + 1, (c + 1) & 1);
            wait_asynccnt<PER_THREAD>();              // chunk c done (in-order)
        } else {
            wait_asynccnt<0>();
        }
        __syncthreads();                              // chunk c visible to all

        if (valid) {
            const float* lBuf = smem + (c & 1) * BUF_LDS;
            const float* lWr  = lBuf + (HAS_AGG ? CK * LDW : 0);
            const int kb = c * CK;
#pragma unroll
            for (int kk = 0; kk < CK; kk += 4) {
                if constexpr (HAS_AGG) {
                    const float* arow = agg + (size_t)(m0 + r) * K + kb + kk + hiK;
                    v2f a; a.x = arow[0] * scale; a.y = arow[1] * scale;
                    const float* lb = lBuf + (kk + hiK) * LDW + j0 + r;
                    v2f b0 = {lb[0],  lb[0  + LDW]};
                    v2f b1 = {lb[16], lb[16 + LDW]};
                    v2f b2 = {lb[32], lb[32 + LDW]};
                    v2f b3 = {lb[48], lb[48 + LDW]};
                    c0 = __builtin_amdgcn_wmma_f32_16x16x4_f32(false, a, false, b0, (short)0, c0, false, false);
                    c1 = __builtin_amdgcn_wmma_f32_16x16x4_f32(false, a, false, b1, (short)0, c1, false, false);
                    c2 = __builtin_amdgcn_wmma_f32_16x16x4_f32(false, a, false, b2, (short)0, c2, false, false);
                    c3 = __builtin_amdgcn_wmma_f32_16x16x4_f32(false, a, false, b3, (short)0, c3, false, false);
                }
                {
                    const float* arow = X + (size_t)(m0 + r) * K + kb + kk + hiK;
                    v2f a; a.x = arow[0]; a.y = arow[1];
                    const float* lb = lWr + (kk + hiK) * LDW + j0 + r;
                    v2f b0 = {lb[0],  lb[0  + LDW]};
                    v2f b1 = {lb[16], lb[16 + LDW]};
                    v2f b2 = {lb[32], lb[32 + LDW]};
                    v2f b3 = {lb[48], lb[48 + LDW]};
                    c0 = __builtin_amdgcn_wmma_f32_16x16x4_f32(false, a, false, b0, (short)0, c0, false, false);
                    c1 = __builtin_amdgcn_wmma_f32_16x16x4_f32(false, a, false, b1, (short)0, c1, false, false);
                    c2 = __builtin_amdgcn_wmma_f32_16x16x4_f32(false, a, false, b2, (short)0, c2, false, false);
                    c3 = __builtin_amdgcn_wmma_f32_16x16x4_f32(false, a, false, b3, (short)0, c3, false, false);
                }
            }
        }
        __syncthreads();                              // buffer reuse safe
    }

    if (!valid) return;

    // Epilogue: bias (+ReLU) and store per C layout.
    const int mbase = m0 + ((lane >> 4) << 3);
    const int col   = j0 + r;
    const float bb0 = bias[col +  0];
    const float bb1 = bias[col + 16];
    const float bb2 = bias[col + 32];
    const float bb3 = bias[col + 48];
#pragma unroll
    for (int i = 0; i < 8; ++i) {
        float v0 = c0[i] + bb0;
        float v1 = c1[i] + bb1;
        float v2 = c2[i] + bb2;
        float v3 = c3[i] + bb3;
        if (doRelu) {
            v0 = fmaxf(v0, 0.0f);
            v1 = fmaxf(v1, 0.0f);
            v2 = fmaxf(v2, 0.0f);
            v3 = fmaxf(v3, 0.0f);
        }
        float* o = out + (size_t)(mbase + i) * NOUT + col;
        o[0]  = v0;
        o[16] = v1;
        o[32] = v2;
        o[48] = v3;
    }
}

// ---------------------------------------------------------------------------
extern "C" void kernel_launch(void* const* d_in, const int* in_sizes, int n_in,
                              void* d_out, int out_size, void* d_ws, size_t ws_size,
                              hipStream_t stream) {
    const float*     x     = (const float*)d_in[0];
    const long long* ei    = (const long long*)d_in[1];
    const float*     W_l1  = (const float*)d_in[2];
    const float*     b_l1  = (const float*)d_in[3];
    const float*     W_r1  = (const float*)d_in[4];
    const float*     W_l2  = (const float*)d_in[5];
    const float*     b_l2  = (const float*)d_in[6];
    const float*     W_r2  = (const float*)d_in[7];
    const float*     W_out = (const float*)d_in[8];
    const float*     b_out = (const float*)d_in[9];

    const int N = NNODES;
    const int E = in_sizes[1] / 2;
    const long long* src = ei;
    const long long* dst = ei + E;

    // Workspace layout (fp32): cnt[N] | agg[N*HID] | h1[N*HID] | h2[N*HID]
    float* cnt = (float*)d_ws;
    float* agg = cnt + N;
    float* h1  = agg + (size_t)N * HID;
    float* h2  = h1  + (size_t)N * HID;

    // ---- degree counts (once) + layer-1 aggregation ----
    hipMemsetAsync(cnt, 0, (size_t)N * sizeof(float), stream);
    hipMemsetAsync(agg, 0, (size_t)N * DIN * sizeof(float), stream);
    count_edges_kernel<<<(E + 255) / 256, 256, 0, stream>>>(dst, cnt, E);
    {
        long long total = (long long)E * (DIN / 4);
        int blocks = (int)((total + 255) / 256);
        scatter_add4_kernel<<<blocks, 256, 0, stream>>>(x, src, dst, agg, E, DIN);
    }

    // ---- layer 1: h1 = relu(agg/cnt @ W_l1 + b_l1 + x @ W_r1) ----
    {
        int blocks = (N + 63) / 64;
        size_t shmem = 2ull * 2ull * 16ull * (HID + 8) * sizeof(float); // 34.8 KB
        sage_gemm_wmma_kernel<HID, DIN, true><<<blocks, 256, shmem, stream>>>(
            agg, cnt, x, W_l1, W_r1, b_l1, h1, N, 1);
    }

    // ---- layer-2 aggregation ----
    hipMemsetAsync(agg, 0, (size_t)N * HID * sizeof(float), stream);
    {
        long long total = (long long)E * (HID / 4);
        int blocks = (int)((total + 255) / 256);
        scatter_add4_kernel<<<blocks, 256, 0, stream>>>(h1, src, dst, agg, E, HID);
    }

    // ---- layer 2: h2 = relu(agg/cnt @ W_l2 + b_l2 + h1 @ W_r2) ----
    {
        int blocks = (N + 63) / 64;
        size_t shmem = 2ull * 2ull * 16ull * (HID + 8) * sizeof(float); // 34.8 KB
        sage_gemm_wmma_kernel<HID, HID, true><<<blocks, 256, shmem, stream>>>(
            agg, cnt, h1, W_l2, W_r2, b_l2, h2, N, 1);
    }

    // ---- output head: out = h2 @ W_out + b_out (no agg phase, no relu) ----
    {
        int blocks = (N + 127) / 128;
        size_t shmem = 2ull * 1ull * 16ull * (DOUT + 8) * sizeof(float); // 9.2 KB
        sage_gemm_wmma_kernel<DOUT, HID, false><<<blocks, 256, shmem, stream>>>(
            nullptr, cnt, h2, W_out, W_out, b_out, (float*)d_out, N, 0);
    }
}